// WayfinderAttentionTorch_31868657337025
// MI455X (gfx1250) — compile-verified
//
#include <hip/hip_runtime.h>
#include <hip/hip_bf16.h>

typedef __attribute__((ext_vector_type(2))) float v2f;
typedef __attribute__((ext_vector_type(4))) float v4f;
typedef __attribute__((ext_vector_type(8))) float v8f;

#define T_TOK 2048
#define C_DIM 1024
#define H_HEADS 16
#define DH_DIM 64
#define D_NEI 64
#define C3 3072

// ---------------------------------------------------------------------------
// GEMM: OUT[m,n] = sum_k A[m,k] * Bw[n,k]   (A: [M,K] row-major, Bw: [N,K] row-major)
//
// Wave tile: 16 (M) x 128 (N), 8 f32 WMMA accumulator chains, K processed in
// chunks of 8 via two V_WMMA_F32_16X16X4_F32 per chain per chunk.
//
// K-column remap trick: WMMA sums its 4 K slots regardless of which physical
// columns we bind to them. We bind wmma-step-0 to columns {k+4h, k+4h+1} and
// wmma-step-1 to {k+4h+2, k+4h+3} (h = lane>>4), identically for A and B, so
// one b128 load per lane per chunk feeds BOTH wmma steps. Result is exact.
//
// Register double-buffering: the 9 loads for chunk k+8 are issued before the
// 16 WMMAs of chunk k, so global loads overlap matrix-pipe execution instead
// of the load->s_wait_loadcnt 0->wmma lockstep of the previous version.
//
// Block = 256 threads = 8 waves (wave32), each wave a distinct M-tile.
// grid.x = N/128, grid.y = M/128.
// ---------------------------------------------------------------------------
__global__ __launch_bounds__(256) void gemm_xt_wmma(const float* __restrict__ A,
                                                    const float* __restrict__ Bw,
                                                    float* __restrict__ Cout,
                                                    int M, int N, int K) {
  const int lane = threadIdx.x & 31;
  const int half = lane >> 4;   // K-slot group this lane holds
  const int lq   = lane & 15;   // M (for A) / N (for B) index within tile
  const int wave = threadIdx.x >> 5;

  const int m0 = (blockIdx.y * 8 + wave) * 16;
  const int n0 = blockIdx.x * 128;

  const float* arow = A  + (size_t)(m0 + lq) * K + 4 * half;
  const float* brow = Bw + (size_t)(n0 + lq) * K + 4 * half;
  const size_t bstride = (size_t)16 * K;  // 16 N-rows between the 8 sub-tiles

  v8f acc[8] = {};

  // prologue: load chunk 0
  v4f a_cur = *(const v4f*)(arow);
  v4f b_cur[8];
#pragma unroll
  for (int i = 0; i < 8; ++i) b_cur[i] = *(const v4f*)(brow + (size_t)i * bstride);

  for (int k = 0; k < K - 8; k += 8) {
    // issue next-chunk loads first so they overlap this chunk's WMMAs
    v4f a_nxt = *(const v4f*)(arow + k + 8);
    v4f b_nxt[8];
#pragma unroll
    for (int i = 0; i < 8; ++i)
      b_nxt[i] = *(const v4f*)(brow + (size_t)i * bstride + k + 8);

    const v2f a0 = {a_cur.x, a_cur.y};
    const v2f a1 = {a_cur.z, a_cur.w};
#pragma unroll
    for (int i = 0; i < 8; ++i) {
      const v2f b0 = {b_cur[i].x, b_cur[i].y};
      acc[i] = __builtin_amdgcn_wmma_f32_16x16x4_f32(
          false, a0, false, b0, (short)0, acc[i], false, false);
    }
#pragma unroll
    for (int i = 0; i < 8; ++i) {
      const v2f b1 = {b_cur[i].z, b_cur[i].w};
      acc[i] = __builtin_amdgcn_wmma_f32_16x16x4_f32(
          false, a1, false, b1, (short)0, acc[i], false, false);
    }

    a_cur = a_nxt;
#pragma unroll
    for (int i = 0; i < 8; ++i) b_cur[i] = b_nxt[i];
  }

  // epilogue: last chunk (no more loads)
  {
    const v2f a0 = {a_cur.x, a_cur.y};
    const v2f a1 = {a_cur.z, a_cur.w};
#pragma unroll
    for (int i = 0; i < 8; ++i) {
      const v2f b0 = {b_cur[i].x, b_cur[i].y};
      acc[i] = __builtin_amdgcn_wmma_f32_16x16x4_f32(
          false, a0, false, b0, (short)0, acc[i], false, false);
    }
#pragma unroll
    for (int i = 0; i < 8; ++i) {
      const v2f b1 = {b_cur[i].z, b_cur[i].w};
      acc[i] = __builtin_amdgcn_wmma_f32_16x16x4_f32(
          false, a1, false, b1, (short)0, acc[i], false, false);
    }
  }

  // C/D frag layout: VGPR r -> row m0 + r + 8*half, lanes 0-15 -> cols n0+i*16+lq
  float* crow = Cout + (size_t)(m0 + 8 * half) * N + n0 + lq;
#pragma unroll
  for (int r = 0; r < 8; ++r) {
#pragma unroll
    for (int i = 0; i < 8; ++i) {
      crow[(size_t)r * N + i * 16] = acc[i][r];
    }
  }
}

// ---------------------------------------------------------------------------
// Gather attention: one wave32 per (h, t).
//  - q[t] staged in per-wave LDS
//  - each lane scores 2 neighbors (float4 K-gathers, all L2-resident)
//  - softmax via shfl_xor wave reductions
//  - weights + safe indices staged in LDS -> wave-uniform broadcast reads in
//    the ctx accumulation loop (uniform branch skips masked neighbors)
// ---------------------------------------------------------------------------
__global__ __launch_bounds__(256) void wayfinder_attn(const float* __restrict__ qkv,
                                                      const int*   __restrict__ neigh_idx,
                                                      const int*   __restrict__ edge_type,
                                                      const float* __restrict__ edge_bias,
                                                      float* __restrict__ ctx) {
  __shared__ __align__(16) float q_s[8][DH_DIM];
  __shared__ float w_s[8][D_NEI];
  __shared__ int   i_s[8][D_NEI];

  const int lane = threadIdx.x & 31;
  const int wave = threadIdx.x >> 5;
  const int wg   = blockIdx.x * 8 + wave;
  const int h    = wg >> 11;            // / T_TOK
  const int t    = wg & (T_TOK - 1);

  const float b1 = edge_bias[0], b2 = edge_bias[1];
  const float b3 = edge_bias[2], b4 = edge_bias[3];

  // stage q[t] for this head into LDS (wave-private slice, no barrier needed)
  const float* qrow = qkv + (size_t)t * C3 + h * DH_DIM;
  q_s[wave][lane]      = qrow[lane];
  q_s[wave][lane + 32] = qrow[lane + 32];

  const size_t nbase = ((size_t)h * T_TOK + t) * D_NEI;
  const int idx0 = neigh_idx[nbase + lane];
  const int idx1 = neigh_idx[nbase + lane + 32];
  const int et0  = edge_type[nbase + lane];
  const int et1  = edge_type[nbase + lane + 32];
  const bool msk0 = (idx0 >= 0) && (idx0 <= t);
  const bool msk1 = (idx1 >= 0) && (idx1 <= t);
  const int safe0 = min(max(idx0, 0), T_TOK - 1);
  const int safe1 = min(max(idx1, 0), T_TOK - 1);

  const float* k0 = qkv + (size_t)safe0 * C3 + C_DIM + h * DH_DIM;
  const float* k1 = qkv + (size_t)safe1 * C3 + C_DIM + h * DH_DIM;

  float s0 = 0.0f, s1 = 0.0f;
#pragma unroll
  for (int d = 0; d < DH_DIM; d += 4) {
    v4f qv  = *(const v4f*)&q_s[wave][d];
    v4f k0v = *(const v4f*)(k0 + d);
    v4f k1v = *(const v4f*)(k1 + d);
    s0 = fmaf(qv.x, k0v.x, s0); s0 = fmaf(qv.y, k0v.y, s0);
    s0 = fmaf(qv.z, k0v.z, s0); s0 = fmaf(qv.w, k0v.w, s0);
    s1 = fmaf(qv.x, k1v.x, s1); s1 = fmaf(qv.y, k1v.y, s1);
    s1 = fmaf(qv.z, k1v.z, s1); s1 = fmaf(qv.w, k1v.w, s1);
  }

  const float fb0 = (et0 == 0) ? 0.0f : (et0 == 1 ? b1 : et0 == 2 ? b2 : et0 == 3 ? b3 : b4);
  const float fb1 = (et1 == 0) ? 0.0f : (et1 == 1 ? b1 : et1 == 2 ? b2 : et1 == 3 ? b3 : b4);

  s0 = s0 * 0.125f + fb0;   // 1/sqrt(64)
  s1 = s1 * 0.125f + fb1;
  s0 = msk0 ? s0 : -1e30f;
  s1 = msk1 ? s1 : -1e30f;

  // wave32 max reduction
  float mx = fmaxf(s0, s1);
#pragma unroll
  for (int off = 16; off; off >>= 1) mx = fmaxf(mx, __shfl_xor(mx, off, 32));

  const float e0 = msk0 ? expf(s0 - mx) : 0.0f;
  const float e1 = msk1 ? expf(s1 - mx) : 0.0f;

  float sum = e0 + e1;
#pragma unroll
  for (int off = 16; off; off >>= 1) sum += __shfl_xor(sum, off, 32);

  const float inv = 1.0f / fmaxf(sum, 1e-20f);
  w_s[wave][lane]      = e0 * inv;
  w_s[wave][lane + 32] = e1 * inv;
  i_s[wave][lane]      = safe0;
  i_s[wave][lane + 32] = safe1;

  // ctx accumulation: lane owns dims (lane, lane+32); weight reads are
  // wave-uniform LDS broadcasts, so the w!=0 branch is a uniform skip.
  float c0 = 0.0f, c1 = 0.0f;
  const float* vbase = qkv + 2 * C_DIM + h * DH_DIM;
  for (int n = 0; n < D_NEI; ++n) {
    const float w = w_s[wave][n];
    if (w != 0.0f) {
      const float* vr = vbase + (size_t)i_s[wave][n] * C3;
      c0 = fmaf(w, vr[lane], c0);
      c1 = fmaf(w, vr[lane + 32], c1);
    }
  }

  float* crow = ctx + (size_t)t * C_DIM + h * DH_DIM;
  crow[lane]      = c0;
  crow[lane + 32] = c1;
}

extern "C" void kernel_launch(void* const* d_in, const int* in_sizes, int n_in,
                              void* d_out, int out_size, void* d_ws, size_t ws_size,
                              hipStream_t stream) {
  const float* x         = (const float*)d_in[0];   // [T, C]
  const float* Wqkv      = (const float*)d_in[1];   // [3C, C]
  const float* Wout      = (const float*)d_in[2];   // [C, C]
  const float* edge_bias = (const float*)d_in[3];   // [4]
  const int*   neigh_idx = (const int*)d_in[4];     // [H, T, D]
  const int*   edge_type = (const int*)d_in[5];     // [H, T, D]
  float* out = (float*)d_out;                       // [T, C]

  float* qkv = (float*)d_ws;                        // [T, 3C]  (q|k|v per row)
  float* ctx = qkv + (size_t)T_TOK * C3;            // [T, C]

  dim3 blk(256);

  // 1) qkv = x @ Wqkv^T   (M=2048, N=3072, K=1024)
  gemm_xt_wmma<<<dim3(C3 / 128, T_TOK / 128), blk, 0, stream>>>(
      x, Wqkv, qkv, T_TOK, C3, C_DIM);

  // 2) gathered-neighbor attention -> ctx [T, C]
  wayfinder_attn<<<dim3((H_HEADS * T_TOK) / 8), blk, 0, stream>>>(
      qkv, neigh_idx, edge_type, edge_bias, ctx);

  // 3) out = ctx @ Wout^T  (M=2048, N=1024, K=1024)
  gemm_xt_wmma<<<dim3(C_DIM / 128, T_TOK / 128), blk, 0, stream>>>(
      ctx, Wout, out, T_TOK, C_DIM, C_DIM);
}